// GCNup_57501022159518
// MI455X (gfx1250) — compile-verified
//
#include <hip/hip_runtime.h>
#include <hip/hip_bf16.h>

typedef __attribute__((ext_vector_type(2))) float v2f;
typedef __attribute__((ext_vector_type(8))) float v8f;

#define HID 128
#define F_IN 25

// ---------------- degree / normalization ----------------
__global__ void k_init_deg(float* deg, int n) {
  int i = blockIdx.x * blockDim.x + threadIdx.x;
  if (i < n) deg[i] = 1.0f;  // self-loop contributes 1 to every target degree
}

__global__ void k_count_deg(const int* __restrict__ col, float* deg, int e) {
  int i = blockIdx.x * blockDim.x + threadIdx.x;
  if (i < e) atomicAdd(&deg[col[i]], 1.0f);
}

__global__ void k_rsqrt_inplace(float* deg, int n) {
  int i = blockIdx.x * blockDim.x + threadIdx.x;
  if (i < n) deg[i] = rsqrtf(deg[i]);  // deg >= 1 always (self-loops)
}

__global__ void k_zero(float* p, long long n) {
  long long i = (long long)blockIdx.x * blockDim.x + threadIdx.x;
  if (i < n) p[i] = 0.0f;
}

// C/D tile store: fast path = uniform full tile, one clause of 8 stores.
__device__ __forceinline__ void store_ctile(float* __restrict__ H, int mbase, int ncol,
                                            int M, const v8f& c) {
  float* hp = H + (long long)mbase * HID + ncol;
  if (mbase + 7 < M) {  // uniform across the wave (tile-aligned M hits this always)
#pragma unroll
    for (int r = 0; r < 8; ++r) hp[(long long)r * HID] = c[r];
  } else {
#pragma unroll
    for (int r = 0; r < 8; ++r)
      if (mbase + r < M) hp[(long long)r * HID] = c[r];
  }
}

// ---------------- layer-1 GEMM: H[M,128] = X[M,25] @ W[25,128] ----------------
// One wave -> one 16x16 tile; block = 8 waves -> 16 rows x 128 cols.
// K padded 25->28; tail handled with clamped loads + selects (no EXEC branches).
__global__ void k_gemm_wmma25(const float* __restrict__ X, const float* __restrict__ W,
                              float* __restrict__ H, int M) {
  const int wave = threadIdx.x >> 5;
  const int lane = threadIdx.x & 31;
  const int hl   = lane >> 4;           // 32-bit A layout: half 0 -> K=0,1; half 1 -> K=2,3
  const int l15  = lane & 15;
  const int mrow = (blockIdx.x << 4) + l15;
  const int ncol = (wave << 4) + l15;
  const int mclamp = mrow < M ? mrow : M - 1;  // clamp; bad rows never stored
  const float* xrow = X + (long long)mclamp * F_IN;

  v8f c = {0.f, 0.f, 0.f, 0.f, 0.f, 0.f, 0.f, 0.f};
#pragma unroll
  for (int k0 = 0; k0 < 28; k0 += 4) {
    const int ka = k0 + hl * 2;
    const int i0 = (ka < F_IN) ? ka : 0;
    const int i1 = (ka + 1 < F_IN) ? ka + 1 : 0;
    float a0 = xrow[i0], a1 = xrow[i1];
    float b0 = W[i0 * HID + ncol], b1 = W[i1 * HID + ncol];
    v2f a, b;
    a.x = (ka < F_IN) ? a0 : 0.f;
    a.y = (ka + 1 < F_IN) ? a1 : 0.f;
    b.x = (ka < F_IN) ? b0 : 0.f;
    b.y = (ka + 1 < F_IN) ? b1 : 0.f;
    c = __builtin_amdgcn_wmma_f32_16x16x4_f32(false, a, false, b, (short)0, c,
                                              false, false);
  }
  store_ctile(H, (blockIdx.x << 4) + (hl << 3), ncol, M, c);
}

// ---------------- layer-2 GEMM: H[M,128] = X[M,128] @ W[128,128] ----------------
// W (64 KB) staged into LDS with CDNA5 async global->LDS loads (ASYNCcnt),
// then all 8 waves feed B-fragments from LDS. No guards anywhere in the k-loop.
__global__ void k_gemm_wmma128(const float* __restrict__ X, const float* __restrict__ W,
                               float* __restrict__ H, int M) {
  __shared__ float sW[128 * HID];  // 64 KB of the 320 KB WGP LDS

  const int tid = threadIdx.x;
  // 4096 16-byte chunks, 256 threads -> 16 async copies per thread
#pragma unroll
  for (int i = 0; i < 16; ++i) {
    const int idx4 = tid + i * 256;                     // float4 index
    unsigned lds_addr = (unsigned)(size_t)(&sW[idx4 * 4]);
    unsigned long long gaddr = (unsigned long long)(const void*)(W + idx4 * 4);
    asm volatile("global_load_async_to_lds_b128 %0, %1, off"
                 :: "v"(lds_addr), "v"(gaddr) : "memory");
  }
  asm volatile("s_wait_asynccnt 0x0" ::: "memory");
  __syncthreads();

  const int wave = tid >> 5;
  const int lane = tid & 31;
  const int hl   = lane >> 4;
  const int l15  = lane & 15;
  const int mrow = (blockIdx.x << 4) + l15;
  const int ncol = (wave << 4) + l15;
  const int mclamp = mrow < M ? mrow : M - 1;
  const float* xrow = X + (long long)mclamp * HID;  // row stride 512 B -> b64 aligned

  v8f c = {0.f, 0.f, 0.f, 0.f, 0.f, 0.f, 0.f, 0.f};
#pragma unroll
  for (int k0 = 0; k0 < 128; k0 += 4) {
    const int ka = k0 + hl * 2;           // even
    v2f a = *(const v2f*)(xrow + ka);     // global_load_b64
    v2f b;
    b.x = sW[ka * HID + ncol];            // ds_load
    b.y = sW[(ka + 1) * HID + ncol];
    c = __builtin_amdgcn_wmma_f32_16x16x4_f32(false, a, false, b, (short)0, c,
                                              false, false);
  }
  store_ctile(H, (blockIdx.x << 4) + (hl << 3), ncol, M, c);
}

// ---------------- edge scatter (message passing) ----------------
// One wave per edge; each lane carries 4 features (128 = 32 lanes * 4).
__global__ void k_scatter(const float* __restrict__ H, const int* __restrict__ row,
                          const int* __restrict__ col, const float* __restrict__ dinv,
                          float* __restrict__ out, int E) {
  long long t = (long long)blockIdx.x * blockDim.x + threadIdx.x;
  int e = (int)(t >> 5);
  if (e >= E) return;
  int lane = threadIdx.x & 31;
  int r = row[e], c = col[e];
  float nrm = dinv[r] * dinv[c];
  float4 v = ((const float4*)(H + (long long)r * HID))[lane];
  float* op = out + (long long)c * HID + lane * 4;
  atomicAdd(op + 0, v.x * nrm);
  atomicAdd(op + 1, v.y * nrm);
  atomicAdd(op + 2, v.z * nrm);
  atomicAdd(op + 3, v.w * nrm);
}

// out = relu(out + H*dinv^2 (self-loop) + bias)
__global__ void k_epilogue(const float* __restrict__ H, const float* __restrict__ dinv,
                           const float* __restrict__ bias, float* __restrict__ out, int n) {
  long long i = (long long)blockIdx.x * blockDim.x + threadIdx.x;
  if (i >= (long long)n * HID) return;
  int node = (int)(i >> 7);
  int f = (int)(i & (HID - 1));
  float di = dinv[node];
  float v = out[i] + H[i] * (di * di) + bias[f];
  out[i] = fmaxf(v, 0.0f);
}

// ---------------- batchnorm ----------------
__global__ void k_bn_stats(const float* __restrict__ H, float* __restrict__ stats, int n) {
  int f = threadIdx.x;  // blockDim = 128 (thread = feature); grid-stride over rows
  float s = 0.f, s2 = 0.f;
  for (int i = blockIdx.x; i < n; i += gridDim.x) {
    float v = H[(long long)i * HID + f];
    s += v;
    s2 += v * v;
  }
  atomicAdd(&stats[f], s);
  atomicAdd(&stats[HID + f], s2);
}

__global__ void k_bn_apply(float* __restrict__ H, const float* __restrict__ stats,
                           const float* __restrict__ gamma, const float* __restrict__ beta,
                           int n) {
  long long i = (long long)blockIdx.x * blockDim.x + threadIdx.x;
  if (i >= (long long)n * HID) return;
  int f = (int)(i & (HID - 1));
  float inv_n = 1.0f / (float)n;
  float mean = stats[f] * inv_n;
  float var = stats[HID + f] * inv_n - mean * mean;
  float scale = gamma[f] * rsqrtf(var + 1e-5f);
  H[i] = (H[i] - mean) * scale + beta[f];
}

// ---------------- driver ----------------
extern "C" void kernel_launch(void* const* d_in, const int* in_sizes, int n_in,
                              void* d_out, int out_size, void* d_ws, size_t ws_size,
                              hipStream_t stream) {
  const float* x     = (const float*)d_in[0];
  const int*   ei    = (const int*)d_in[1];
  const float* W1    = (const float*)d_in[2];
  const float* b1    = (const float*)d_in[3];
  const float* gamma = (const float*)d_in[4];
  const float* beta  = (const float*)d_in[5];
  const float* W2    = (const float*)d_in[6];
  const float* b2    = (const float*)d_in[7];
  float* out = (float*)d_out;

  const int N = in_sizes[0] / F_IN;   // 100000
  const int E = in_sizes[1] / 2;      // 1600000
  const int* row = ei;                // sources
  const int* col = ei + E;            // targets

  // workspace layout (floats): dinv[N] | bufA[N*128] | bufB[N*128] | stats[256]
  float* dinv  = (float*)d_ws;
  float* bufA  = dinv + N;
  float* bufB  = bufA + (long long)N * HID;
  float* stats = bufB + (long long)N * HID;

  const long long NH = (long long)N * HID;
  auto cdiv = [](long long a, long long b) { return (unsigned)((a + b - 1) / b); };

  // symmetric normalization coefficients
  k_init_deg<<<cdiv(N, 256), 256, 0, stream>>>(dinv, N);
  k_count_deg<<<cdiv(E, 256), 256, 0, stream>>>(col, dinv, E);
  k_rsqrt_inplace<<<cdiv(N, 256), 256, 0, stream>>>(dinv, N);

  // ---- layer 1: GCNConv(25 -> 128) + ReLU ----
  k_gemm_wmma25<<<cdiv(N, 16), 256, 0, stream>>>(x, W1, bufA, N);
  k_zero<<<cdiv(NH, 256), 256, 0, stream>>>(bufB, NH);
  k_scatter<<<cdiv((long long)E * 32, 256), 256, 0, stream>>>(bufA, row, col, dinv, bufB, E);
  k_epilogue<<<cdiv(NH, 256), 256, 0, stream>>>(bufA, dinv, b1, bufB, N);

  // ---- BatchNorm1d (training-mode batch stats) ----
  k_zero<<<1, 256, 0, stream>>>(stats, 256);
  k_bn_stats<<<512, HID, 0, stream>>>(bufB, stats, N);
  k_bn_apply<<<cdiv(NH, 256), 256, 0, stream>>>(bufB, stats, gamma, beta, N);

  // ---- layer 2: GCNConv(128 -> 128) + ReLU ----
  k_gemm_wmma128<<<cdiv(N, 16), 256, 0, stream>>>(bufB, W2, bufA, N);
  k_zero<<<cdiv(NH, 256), 256, 0, stream>>>(out, NH);
  k_scatter<<<cdiv((long long)E * 32, 256), 256, 0, stream>>>(bufA, row, col, dinv, out, E);
  k_epilogue<<<cdiv(NH, 256), 256, 0, stream>>>(bufA, dinv, b2, out, N);
}